// agss_14104672600181
// MI455X (gfx1250) — compile-verified
//
#include <hip/hip_runtime.h>

// Problem constants (match reference)
#define B_   128
#define C_   512
#define T_   512
#define K_   13             // Gaussian taps
#define HALO 6              // K_/2
#define TC   32             // timesteps per chunk
#define CT   128            // channels per block (main pass)
#define BROWS (CT + 2*HALO) // 140 staged channel rows (with halo)
#define BPITCH (TC + 1)     // 33 dwords: coprime with 64 banks -> conflict-free

typedef float v2f __attribute__((ext_vector_type(2)));
typedef float v8f __attribute__((ext_vector_type(8)));

// ---------------------------------------------------------------------------
// Pass A: sigma[b,t] = clip(5.5 + dot(x[b,:,t], W), 1, 10) via
// V_WMMA_F32_16X16X4_F32. One wave per 16-t strip; 8 waves/block; 512 blocks.
// B-fragment broadcasts W across all 16 N columns so every D column holds the
// dot product; lanes 0/16 extract column N=0.
// ---------------------------------------------------------------------------
__launch_bounds__(256)
__global__ void sigma_wmma_kernel(const float* __restrict__ x,
                                  const float* __restrict__ W,
                                  float* __restrict__ sigma) {
    __shared__ float Wl[C_];
    const int tid  = threadIdx.x;
    const int lane = tid & 31;
    const int wave = tid >> 5;
    const int b    = blockIdx.y;
    const int t0   = (blockIdx.x * 8 + wave) * 16;   // 32 strips per batch

    for (int i = tid; i < C_; i += 256) Wl[i] = W[i];
    __syncthreads();

    const float* xb = x + (size_t)b * C_ * T_;
    const int m   = lane & 15;
    const int kh  = (lane >> 4) << 1;        // A: lanes0-15 K=0,1 ; lanes16-31 K=2,3
    const int kb0 = (lane < 16) ? 0 : 2;     // B: VGPR0 K=0/2, VGPR1 K=1/3
    const float* ap = xb + t0 + m;           // 16 lanes -> contiguous 64B per channel

    v8f acc = {};
    for (int c0 = 0; c0 < C_; c0 += 4) {
        v2f a, bf;
        a.x  = ap[(c0 + kh) * T_];
        a.y  = ap[(c0 + kh + 1) * T_];
        bf.x = Wl[c0 + kb0];
        bf.y = Wl[c0 + kb0 + 1];
        acc = __builtin_amdgcn_wmma_f32_16x16x4_f32(
                  false, a, false, bf, (short)0, acc, false, false);
    }
    // D layout: VGPR r -> M=r (lanes 0-15) / M=r+8 (lanes 16-31); N = lane&15.
    if (lane == 0 || lane == 16) {
        const int mb = (lane == 16) ? 8 : 0;
        #pragma unroll
        for (int r = 0; r < 8; ++r) {
            float s = 5.5f + acc[r];
            s = fminf(fmaxf(s, 1.0f), 10.0f);
            sigma[(size_t)b * T_ + t0 + mb + r] = s;
        }
    }
}

// ---------------------------------------------------------------------------
// Pass B: per-channel Gaussian blur (13 taps along c) + sequential LIF scan
// over t. Grid (C/CT, B) = 512 blocks x 128 threads (thread = one channel).
// Double-buffered LDS tile filled with GLOBAL_LOAD_ASYNC_TO_LDS_B32: chunk
// i+1 streams HBM->LDS via the async engine while chunk i is computed.
// ---------------------------------------------------------------------------
__launch_bounds__(128)
__global__ void blur_lif_kernel(const float* __restrict__ x,
                                const float* __restrict__ sigma,
                                float* __restrict__ out) {
    __shared__ float btile[2][BROWS][BPITCH];   // [buf][channel row][t + pad]
    __shared__ float k1d[TC][K_ + 1];

    const int tid   = threadIdx.x;
    const int b     = blockIdx.y;
    const int cbase = blockIdx.x * CT;
    const float* xb = x + (size_t)b * C_ * T_;
    float*       ob = out + ((size_t)b * C_ + cbase + tid) * T_;

    // Issue async fill for chunk starting at t0 into buffer `buf`.
    // Lanes sweep tt (contiguous 128B in global, contiguous dwords in LDS).
    auto issue_chunk = [&](int t0, int buf) {
        for (int i = tid; i < BROWS * TC; i += 128) {
            const int cc = i >> 5;            // 0..139
            const int tt = i & 31;
            const int ch = cbase + cc - HALO;
            if (ch >= 0 && ch < C_) {
                const uint32_t ldsoff =
                    (uint32_t)(uintptr_t)&btile[buf][cc][tt];
                const uint64_t gaddr =
                    (uint64_t)(uintptr_t)(xb + (size_t)ch * T_ + t0 + tt);
                asm volatile("global_load_async_to_lds_b32 %0, %1, off"
                             :: "v"(ldsoff), "v"(gaddr) : "memory");
            } else {
                btile[buf][cc][tt] = 0.0f;    // zero-pad channel edges
            }
        }
    };

    issue_chunk(0, 0);
    float mem = 0.0f;                         // LIF state across chunks

    for (int ci = 0; ci < T_ / TC; ++ci) {
        const int t0  = ci * TC;
        const int buf = ci & 1;

        asm volatile("s_wait_asynccnt 0x0" ::: "memory");  // my chunk-ci loads done
        __syncthreads();                                   // tile visible to all

        if (ci + 1 < T_ / TC) issue_chunk(t0 + TC, buf ^ 1);  // overlap next fill

        // Normalized Gaussian taps per t-row (32 threads, 13 v_exp_f32 each)
        if (tid < TC) {
            const float inv = 1.0f / sigma[(size_t)b * T_ + t0 + tid];
            float e[K_];
            float sum = 0.0f;
            #pragma unroll
            for (int k = 0; k < K_; ++k) {
                float d = (float)(k - HALO) * inv;
                e[k] = __expf(-0.5f * d * d);
                sum += e[k];
            }
            const float rs = 1.0f / sum;
            #pragma unroll
            for (int k = 0; k < K_; ++k) k1d[tid][k] = e[k] * rs;
        }
        __syncthreads();

        // Channel blur + sequential LIF over the 32 timesteps of this chunk.
        // Tap reads: lanes consecutive c -> LDS stride 33 dwords, conflict-free.
        float spkv[TC];
        #pragma unroll
        for (int tt = 0; tt < TC; ++tt) {
            float xm = 0.0f;
            #pragma unroll
            for (int k = 0; k < K_; ++k)
                xm += k1d[tt][k] * btile[buf][tid + k][tt];   // channel c-6+k
            const float u     = btile[buf][tid + HALO][tt] - xm;
            const float reset = (mem > 0.5f) ? 1.0f : 0.0f;   // spike(mem_prev-THR)
            mem = 0.95f * mem + u - reset * 0.5f;
            spkv[tt] = (mem > 0.5f) ? 1.0f : 0.0f;
        }

        // Each lane stores its channel's 32 spikes = one contiguous 128B span
        #pragma unroll
        for (int q = 0; q < TC; q += 4) {
            *reinterpret_cast<float4*>(ob + t0 + q) =
                make_float4(spkv[q], spkv[q + 1], spkv[q + 2], spkv[q + 3]);
        }
        __syncthreads();   // scan done before next iteration reuses buffers
    }
}

extern "C" void kernel_launch(void* const* d_in, const int* in_sizes, int n_in,
                              void* d_out, int out_size, void* d_ws, size_t ws_size,
                              hipStream_t stream) {
    const float* inp = (const float*)d_in[0];   // (B,1,C,T) f32
    const float* W   = (const float*)d_in[1];   // (1,C)     f32
    float*       out = (float*)d_out;           // (B,1,C,T) f32
    float*       sig = (float*)d_ws;            // B*T floats = 256 KB scratch
    (void)in_sizes; (void)n_in; (void)out_size; (void)ws_size;

    sigma_wmma_kernel<<<dim3(T_ / 16 / 8, B_), dim3(256), 0, stream>>>(inp, W, sig);
    blur_lif_kernel<<<dim3(C_ / CT, B_), dim3(128), 0, stream>>>(inp, sig, out);
}